// PointNet2Classification_67070209294485
// MI455X (gfx1250) — compile-verified
//
#include <hip/hip_runtime.h>

typedef __attribute__((ext_vector_type(2))) float v2f;
typedef __attribute__((ext_vector_type(8))) float v8f;

#define NBATCH 16
#define NPTS0  32768

__host__ __device__ constexpr int imax(int a, int b) { return a > b ? a : b; }

// ------------------------------------------------------------------
// Farthest point sampling. One block per batch. Running min-distance
// array lives in dynamic LDS (n*4 bytes; 128KB for n=32768, fits the
// 320KB/WGP CDNA5 LDS). Argmax: shuffle reduce in-wave (wave32), then
// LDS across waves; tie-break = lowest index (matches jnp.argmax).
// ------------------------------------------------------------------
__global__ void k_fps(const float* __restrict__ src, int n, int stride, int npoint,
                      float* __restrict__ out_xyz) {
  extern __shared__ float dist[];
  __shared__ float s_last[3];
  __shared__ float s_red_v[32];
  __shared__ int   s_red_i[32];
  const int b = blockIdx.x;
  const float* base = src + (size_t)b * n * stride;
  const int tid = threadIdx.x, nth = blockDim.x;
  for (int i = tid; i < n; i += nth) dist[i] = 1e10f;
  if (tid == 0) {
    s_last[0] = base[0]; s_last[1] = base[1]; s_last[2] = base[2];
    float* o = out_xyz + (size_t)b * npoint * 3;
    o[0] = base[0]; o[1] = base[1]; o[2] = base[2];
  }
  __syncthreads();
  for (int s = 1; s < npoint; ++s) {
    const float lx = s_last[0], ly = s_last[1], lz = s_last[2];
    float bestd = -1.0f; int besti = 0;
    for (int i = tid; i < n; i += nth) {
      const float* p = base + (size_t)i * stride;
      float dx = p[0] - lx, dy = p[1] - ly, dz = p[2] - lz;
      float d = dx * dx + dy * dy + dz * dz;
      float m = fminf(dist[i], d);
      dist[i] = m;
      if (m > bestd) { bestd = m; besti = i; }
    }
    for (int o = 16; o > 0; o >>= 1) {
      float od = __shfl_down(bestd, o, 32);
      int   oi = __shfl_down(besti, o, 32);
      if (od > bestd || (od == bestd && oi < besti)) { bestd = od; besti = oi; }
    }
    if ((tid & 31) == 0) { s_red_v[tid >> 5] = bestd; s_red_i[tid >> 5] = besti; }
    __syncthreads();
    if (tid == 0) {
      const int nw = nth >> 5;
      float bd = s_red_v[0]; int bi = s_red_i[0];
      for (int w = 1; w < nw; ++w)
        if (s_red_v[w] > bd || (s_red_v[w] == bd && s_red_i[w] < bi)) { bd = s_red_v[w]; bi = s_red_i[w]; }
      const float* p = base + (size_t)bi * stride;
      s_last[0] = p[0]; s_last[1] = p[1]; s_last[2] = p[2];
      float* o = out_xyz + ((size_t)b * npoint + s) * 3;
      o[0] = p[0]; o[1] = p[1]; o[2] = p[2];
    }
    __syncthreads();
  }
}

// ------------------------------------------------------------------
// Ball query: one wave32 per center. Ordered stream compaction with
// ballot + prefix popcount keeps index order (matches sort-based ref);
// pad tail slots with the first hit. count is wave-uniform.
// ------------------------------------------------------------------
__global__ void k_ballquery(const float* __restrict__ pts, int n, int stride,
                            const float* __restrict__ centers, int S, int nB,
                            float r2, int nsample, int* __restrict__ out_idx) {
  __shared__ int sh_idx[8][64];
  const int wav = threadIdx.x >> 5, lane = threadIdx.x & 31;
  const int c = blockIdx.x * 8 + wav;
  if (c >= nB * S) return;
  const int b = c / S;
  const float cx = centers[(size_t)c * 3 + 0];
  const float cy = centers[(size_t)c * 3 + 1];
  const float cz = centers[(size_t)c * 3 + 2];
  const float* base = pts + (size_t)b * n * stride;
  int count = 0;
  for (int i0 = 0; i0 < n && count < nsample; i0 += 32) {
    const int i = i0 + lane;
    bool pred = false;
    if (i < n) {
      const float* p = base + (size_t)i * stride;
      float dx = p[0] - cx, dy = p[1] - cy, dz = p[2] - cz;
      pred = (dx * dx + dy * dy + dz * dz) < r2;
    }
    const unsigned mask = (unsigned)__ballot(pred);
    const int prefix = __popc(mask & ((1u << lane) - 1u));
    if (pred) {
      const int slot = count + prefix;
      if (slot < nsample) sh_idx[wav][slot] = i;
    }
    count += __popc(mask);
  }
  if (count > nsample) count = nsample;
  const int first = (count > 0) ? sh_idx[wav][0] : 0;   // center itself always hits
  for (int k = count + lane; k < nsample; k += 32) sh_idx[wav][k] = first;
  for (int k = lane; k < nsample; k += 32)
    out_idx[(size_t)c * nsample + k] = sh_idx[wav][k];
}

// ------------------------------------------------------------------
// Gather grouped rows: g[row][c] = (c<3 ? xyz[pt]-center : feats[pt])
// zero-padded to Kpad (multiple of 4) so the f32 WMMA K-loop is clean.
// ------------------------------------------------------------------
__global__ void k_gather(const float* __restrict__ pts, int n, int stride,
                         const float* __restrict__ fsrc, int fstride, int C,
                         const float* __restrict__ centers,
                         const int* __restrict__ idx,
                         int S, int nsample, int Kpad, size_t total,
                         float* __restrict__ g) {
  for (size_t e = (size_t)blockIdx.x * blockDim.x + threadIdx.x; e < total;
       e += (size_t)gridDim.x * blockDim.x) {
    const int c = (int)(e % Kpad);
    const size_t row = e / Kpad;
    const int k = (int)(row % nsample);
    const size_t cen = row / nsample;
    const int b = (int)(cen / S);
    const int pt = idx[cen * nsample + k];
    float v = 0.0f;
    if (c < 3)            v = pts[((size_t)b * n + pt) * stride + c] - centers[cen * 3 + c];
    else if (c < 3 + C)   v = fsrc[((size_t)b * n + pt) * fstride + (c - 3)];
    g[row * Kpad + c] = v;
  }
}

// g3[row][c] = concat(xyz2, f2) rows for the global SA3 stage (no centering).
__global__ void k_build_g3(const float* __restrict__ xyz, const float* __restrict__ feats,
                           int C, int Kpad, size_t total, float* __restrict__ g) {
  for (size_t e = (size_t)blockIdx.x * blockDim.x + threadIdx.x; e < total;
       e += (size_t)gridDim.x * blockDim.x) {
    const int c = (int)(e % Kpad);
    const size_t row = e / Kpad;
    float v = 0.0f;
    if (c < 3)          v = xyz[row * 3 + c];
    else if (c < 3 + C) v = feats[row * C + (c - 3)];
    g[row * Kpad + c] = v;
  }
}

// ------------------------------------------------------------------
// One layer of the fused shared-MLP on V_WMMA_F32_16X16X4_F32, fully
// compile-time shaped: tile div/mod becomes shifts, K-loop unrolls,
// layer-1 zero-pad guard folds away when KTRUE == KPAD.
// A frag (16x4 f32): lanes 0-15 row M, v.x/v.y = K,K+1; lanes 16-31 = K+2,K+3.
// B frag (4x16 f32): symmetric with N on lanes. C/D: VGPR j -> row j (+8 hi half).
// Tile loop bound depends only on wave id -> wave-uniform -> EXEC all ones.
// ------------------------------------------------------------------
template <int M, int KPAD, int KTRUE, int N>
__device__ __forceinline__ void mlp_layer(const float* __restrict__ Xb,
                                          float* __restrict__ Yb,
                                          const float* __restrict__ W,
                                          const float* __restrict__ bias,
                                          int wav, int nwaves, int half, int r) {
  constexpr int Mtiles = M / 16, Ntiles = N / 16;
  for (int t = wav; t < Mtiles * Ntiles; t += nwaves) {
    const int mt = t / Ntiles, nt = t % Ntiles;           // Ntiles: power of two
    const int n = nt * 16 + r;
    v8f acc;
    const float bv = bias[n];
#pragma unroll
    for (int j = 0; j < 8; ++j) acc[j] = bv;
#pragma unroll 4
    for (int kk = 0; kk < KPAD; kk += 4) {
      const int ka = kk + 2 * half;
      v2f a, bf;
      a.x = Xb[(mt * 16 + r) * KPAD + ka];
      a.y = Xb[(mt * 16 + r) * KPAD + ka + 1];
      if (KTRUE == KPAD) {
        bf.x = W[(size_t)ka * N + n];
        bf.y = W[(size_t)(ka + 1) * N + n];
      } else {
        bf.x = (ka     < KTRUE) ? W[(size_t)ka * N + n]       : 0.0f;
        bf.y = (ka + 1 < KTRUE) ? W[(size_t)(ka + 1) * N + n] : 0.0f;
      }
      acc = __builtin_amdgcn_wmma_f32_16x16x4_f32(
          /*neg_a=*/false, a, /*neg_b=*/false, bf,
          /*c_mod=*/(short)0, acc, /*reuse_a=*/false, /*reuse_b=*/false);
    }
#pragma unroll
    for (int j = 0; j < 8; ++j)
      Yb[(mt * 16 + j + 8 * half) * N + n] = fmaxf(acc[j], 0.0f);  // ReLU every layer
  }
}

// Per-group 3-layer MLP + column max. One block (8 waves) per group of M rows.
// Ping-pongs X<->Y in dynamic LDS; final max reduces directly from Yb (the Xb
// buffer is sized for layer *inputs* only).
template <int M, int K0PAD, int K0, int N1, int N2, int N3>
__global__ void __launch_bounds__(256)
k_mlp_max(const float* __restrict__ g,
          const float* __restrict__ w1, const float* __restrict__ b1,
          const float* __restrict__ w2, const float* __restrict__ b2,
          const float* __restrict__ w3, const float* __restrict__ b3,
          float* __restrict__ out) {
  constexpr int XE = imax(M * K0PAD, imax(M * N1, M * N2));
  extern __shared__ float lds[];
  float* Xb = lds;
  float* Yb = lds + XE;
  const int grp = blockIdx.x;
  const int tid = threadIdx.x, nth = blockDim.x;
  const int wav = tid >> 5, lane = tid & 31;
  const int half = lane >> 4, r = lane & 15;
  const int nwaves = nth >> 5;

  const float* src = g + (size_t)grp * M * K0PAD;
  for (int e = tid; e < M * K0PAD; e += nth) Xb[e] = src[e];
  __syncthreads();

  mlp_layer<M, K0PAD, K0, N1>(Xb, Yb, w1, b1, wav, nwaves, half, r);
  __syncthreads();
  for (int e = tid; e < M * N1; e += nth) Xb[e] = Yb[e];
  __syncthreads();

  mlp_layer<M, N1, N1, N2>(Xb, Yb, w2, b2, wav, nwaves, half, r);
  __syncthreads();
  for (int e = tid; e < M * N2; e += nth) Xb[e] = Yb[e];
  __syncthreads();

  mlp_layer<M, N2, N2, N3>(Xb, Yb, w3, b3, wav, nwaves, half, r);
  __syncthreads();

  // max over the M rows -> out[grp][N3], straight from Yb
  for (int n = tid; n < N3; n += nth) {
    float mx = Yb[n];
    for (int m = 1; m < M; ++m) mx = fmaxf(mx, Yb[m * N3 + n]);
    out[(size_t)grp * N3 + n] = mx;
  }
}

// host-side mirror of the kernel's LDS footprint
template <int M, int K0PAD, int N1, int N2, int N3>
constexpr size_t mlp_lds_bytes() {
  constexpr int XE = imax(M * K0PAD, imax(M * N1, M * N2));
  constexpr int YE = imax(M * N1, imax(M * N2, M * N3));
  return (size_t)(XE + YE) * 4;
}

// ------------------------------------------------------------------
// Classifier head: 512->256->256->1 over 16 rows. Tiny; plain VALU.
// ------------------------------------------------------------------
__global__ void __launch_bounds__(256)
k_cls(const float* __restrict__ f3,
      const float* __restrict__ w1, const float* __restrict__ b1,
      const float* __restrict__ w2, const float* __restrict__ b2,
      const float* __restrict__ w3, const float* __restrict__ b3,
      float* __restrict__ out) {
  __shared__ float X[NBATCH * 512];
  __shared__ float Y[NBATCH * 256];
  const int tid = threadIdx.x, nth = blockDim.x;
  for (int e = tid; e < NBATCH * 512; e += nth) X[e] = f3[e];
  __syncthreads();
  for (int e = tid; e < NBATCH * 256; e += nth) {
    const int row = e >> 8, n = e & 255;
    float acc = b1[n];
    for (int k = 0; k < 512; ++k) acc += X[row * 512 + k] * w1[(size_t)k * 256 + n];
    Y[e] = fmaxf(acc, 0.0f);
  }
  __syncthreads();
  for (int e = tid; e < NBATCH * 256; e += nth) X[e] = Y[e];
  __syncthreads();
  for (int e = tid; e < NBATCH * 256; e += nth) {
    const int row = e >> 8, n = e & 255;
    float acc = b2[n];
    for (int k = 0; k < 256; ++k) acc += X[row * 256 + k] * w2[(size_t)k * 256 + n];
    Y[e] = fmaxf(acc, 0.0f);
  }
  __syncthreads();
  if (tid < NBATCH) {
    float acc = b3[0];
    for (int k = 0; k < 256; ++k) acc += Y[tid * 256 + k] * w3[k];
    out[tid] = acc;                 // no ReLU on final layer
  }
}

// ------------------------------------------------------------------
extern "C" void kernel_launch(void* const* d_in, const int* in_sizes, int n_in,
                              void* d_out, int out_size, void* d_ws, size_t ws_size,
                              hipStream_t stream) {
  const float* pc = (const float*)d_in[0];
  const float* sa1_w[3] = {(const float*)d_in[1],  (const float*)d_in[2],  (const float*)d_in[3]};
  const float* sa1_b[3] = {(const float*)d_in[4],  (const float*)d_in[5],  (const float*)d_in[6]};
  const float* sa2_w[3] = {(const float*)d_in[7],  (const float*)d_in[8],  (const float*)d_in[9]};
  const float* sa2_b[3] = {(const float*)d_in[10], (const float*)d_in[11], (const float*)d_in[12]};
  const float* sa3_w[3] = {(const float*)d_in[13], (const float*)d_in[14], (const float*)d_in[15]};
  const float* sa3_b[3] = {(const float*)d_in[16], (const float*)d_in[17], (const float*)d_in[18]};
  const float* cls_w[3] = {(const float*)d_in[19], (const float*)d_in[20], (const float*)d_in[21]};
  const float* cls_b[3] = {(const float*)d_in[22], (const float*)d_in[23], (const float*)d_in[24]};

  char* base = (char*)d_ws;
  size_t off = 0;
  auto alloc = [&](size_t bytes) -> void* {
    void* p = base + off;
    off = (off + bytes + 255) & ~(size_t)255;
    return p;
  };
  float* nxyz1 = (float*)alloc((size_t)NBATCH * 128 * 3 * 4);
  int*   idx1  = (int*)  alloc((size_t)NBATCH * 128 * 64 * 4);
  float* g1    = (float*)alloc((size_t)NBATCH * 128 * 64 * 8 * 4);     // Kpad=8
  float* f1    = (float*)alloc((size_t)NBATCH * 128 * 128 * 4);
  float* nxyz2 = (float*)alloc((size_t)NBATCH * 32 * 3 * 4);
  int*   idx2  = (int*)  alloc((size_t)NBATCH * 32 * 64 * 4);
  float* g2    = (float*)alloc((size_t)NBATCH * 32 * 64 * 132 * 4);    // Kpad=132
  float* f2    = (float*)alloc((size_t)NBATCH * 32 * 256 * 4);
  float* g3    = (float*)alloc((size_t)NBATCH * 32 * 260 * 4);         // Kpad=260
  float* f3    = (float*)alloc((size_t)NBATCH * 512 * 4);

  // ---- SA1 ----
  k_fps<<<NBATCH, 512, NPTS0 * 4, stream>>>(pc, NPTS0, 6, 128, nxyz1);
  k_ballquery<<<(NBATCH * 128) / 8, 256, 0, stream>>>(pc, NPTS0, 6, nxyz1, 128, NBATCH,
                                                      0.2f * 0.2f, 64, idx1);
  {
    size_t total = (size_t)NBATCH * 128 * 64 * 8;
    k_gather<<<2048, 256, 0, stream>>>(pc, NPTS0, 6, pc + 3, 6, 3, nxyz1, idx1,
                                       128, 64, 8, total, g1);
  }
  k_mlp_max<64, 8, 6, 128, 128, 128>
      <<<NBATCH * 128, 256, mlp_lds_bytes<64, 8, 128, 128, 128>(), stream>>>(
          g1, sa1_w[0], sa1_b[0], sa1_w[1], sa1_b[1], sa1_w[2], sa1_b[2], f1);

  // ---- SA2 ----
  k_fps<<<NBATCH, 128, 128 * 4, stream>>>(nxyz1, 128, 3, 32, nxyz2);
  k_ballquery<<<(NBATCH * 32) / 8, 256, 0, stream>>>(nxyz1, 128, 3, nxyz2, 32, NBATCH,
                                                     0.4f * 0.4f, 64, idx2);
  {
    size_t total = (size_t)NBATCH * 32 * 64 * 132;
    k_gather<<<4096, 256, 0, stream>>>(nxyz1, 128, 3, f1, 128, 128, nxyz2, idx2,
                                       32, 64, 132, total, g2);
  }
  k_mlp_max<64, 132, 131, 128, 128, 256>
      <<<NBATCH * 32, 256, mlp_lds_bytes<64, 132, 128, 128, 256>(), stream>>>(
          g2, sa2_w[0], sa2_b[0], sa2_w[1], sa2_b[1], sa2_w[2], sa2_b[2], f2);

  // ---- SA3 (global) ----
  {
    size_t total = (size_t)NBATCH * 32 * 260;
    k_build_g3<<<256, 256, 0, stream>>>(nxyz2, f2, 256, 260, total, g3);
  }
  k_mlp_max<32, 260, 259, 256, 256, 512>
      <<<NBATCH, 256, mlp_lds_bytes<32, 260, 256, 256, 512>(), stream>>>(
          g3, sa3_w[0], sa3_b[0], sa3_w[1], sa3_b[1], sa3_w[2], sa3_b[2], f3);

  // ---- classifier ----
  k_cls<<<1, 256, 0, stream>>>(f3, cls_w[0], cls_b[0], cls_w[1], cls_b[1],
                               cls_w[2], cls_b[2], (float*)d_out);
}